// Net_395136991234
// MI455X (gfx1250) — compile-verified
//
#include <hip/hip_runtime.h>
#include <hip/hip_bf16.h>
#include <stdint.h>

// Problem sizes (fixed by the reference).
#define IN_DIM  8192
#define HID_DIM 8192
#define OUT_DIM 1024
#define NIN     8192   // n_in is 8192 for both layers

typedef __attribute__((ext_vector_type(2))) float v2f;
typedef __attribute__((ext_vector_type(8))) float v8f;
typedef int v4i_vs __attribute__((vector_size(16)));   // matches builtin param type

#define GLOBAL_AS __attribute__((address_space(1)))
#define LDS_AS    __attribute__((address_space(3)))

// ---- CDNA5 async global->LDS staging (ASYNCcnt path), guarded ----
__device__ __forceinline__ void async_g2l_b128(const float* g, float* l) {
#if defined(__HIP_DEVICE_COMPILE__) && __has_builtin(__builtin_amdgcn_global_load_async_to_lds_b128)
  __builtin_amdgcn_global_load_async_to_lds_b128(
      (GLOBAL_AS v4i_vs*)(uintptr_t)g,
      (LDS_AS v4i_vs*)(uint32_t)(uintptr_t)l,  // low 32 bits of generic addr = LDS byte offset
      0, 0);
#else
  *(float4*)l = *(const float4*)g;             // synchronous fallback
#endif
}

template <int N>
__device__ __forceinline__ void wait_async() {
#if defined(__HIP_DEVICE_COMPILE__) && __has_builtin(__builtin_amdgcn_s_wait_asynccnt)
  __builtin_amdgcn_s_wait_asynccnt((short)N);
#endif // fallback path is synchronous; normal dep tracking + barriers suffice
}

// ============================================================================
// Prep kernel: z = (doExp ? exp(in) : in); scal = {Zsum, sum excluding argmax,
// z[argmax], argmax}.  argmax ties resolve to the LARGEST index, matching the
// stable-argsort semantics of the reference (last element in sorted order).
// Sum-excluding-max is computed exactly (not Zsum - zmax) so a single +inf in
// z1 still yields a finite Zc, as the reference's cumsum does.
// ============================================================================
__global__ __launch_bounds__(256)
void prep_kernel(const float* __restrict__ in, float* __restrict__ zout,
                 float* __restrict__ scal, int n, int doExp) {
  __shared__ float sSum[256];
  __shared__ unsigned long long sMax[256];
  __shared__ int sj;
  const int t = threadIdx.x;

  float ps = 0.f; unsigned long long pm = 0ull;
  for (int j = t; j < n; j += 256) {
    float v = in[j];
    if (doExp) { v = expf(v); zout[j] = v; }
    ps += v;
    unsigned b = __float_as_uint(v);
    unsigned key = (b & 0x80000000u) ? ~b : (b | 0x80000000u); // order-preserving
    unsigned long long pk = ((unsigned long long)key << 32) | (unsigned)j;
    if (pk > pm) pm = pk;
  }
  sSum[t] = ps; sMax[t] = pm;
  __syncthreads();
  for (int s = 128; s; s >>= 1) {
    if (t < s) { sSum[t] += sSum[t + s]; if (sMax[t + s] > sMax[t]) sMax[t] = sMax[t + s]; }
    __syncthreads();
  }
  if (t == 0) { scal[0] = sSum[0]; sj = (int)(sMax[0] & 0xffffffffu); }
  __syncthreads();
  const int jmax = sj;

  float pe = 0.f;
  for (int j = t; j < n; j += 256) {
    if (j == jmax) continue;
    pe += doExp ? zout[j] : in[j];
  }
  sSum[t] = pe;
  __syncthreads();
  for (int s = 128; s; s >>= 1) {
    if (t < s) sSum[t] += sSum[t + s];
    __syncthreads();
  }
  if (t == 0) {
    scal[1] = sSum[0];
    scal[2] = doExp ? zout[jmax] : in[jmax];
    ((int*)scal)[3] = jmax;
  }
}

// ============================================================================
// Layer kernel: 16 output rows per workgroup, 256 threads (8 wave32).
// Pass 1: stream W row-block through LDS (async, double-buffered) and form the
//   16 row sums with V_WMMA_F32_16X16X4_F32 (B = ones => row sums replicated).
// Then: per-row threshold tmp, count L over z (resident in LDS), and resolve:
//   full-sum case / inf case directly; rare masked case re-reads the single
//   needy row (L2-warm) with LDS-atomic reductions.
// ============================================================================
#define ROWS 16
#define KC   128
#define LDW  (KC + 4)     // +4 floats: 16B-aligned pad, rotates banks by 4/row

__global__ __launch_bounds__(256)
void spiking_layer_kernel(const float* __restrict__ W,   // [n_out, NIN]
                          const float* __restrict__ z,   // [NIN]
                          const float* __restrict__ scal,// Zsum, ZexclMax, zmax, jmax
                          float* __restrict__ out) {
  __shared__ float zs[NIN];                 // 32 KB
  __shared__ float tile[2][ROWS * LDW];     // ~16.5 KB
  __shared__ float redW[8][ROWS];
  __shared__ float sWsum[ROWS], sTmp[ROWS];
  __shared__ int   sCnt[ROWS];
  __shared__ int   sNeedyN, sNeedy[ROWS];
  __shared__ float sMW, sMZ;
  __shared__ unsigned long long sMM;

  const int t = threadIdx.x, lane = t & 31, wave = t >> 5;
  const int row0 = blockIdx.x * ROWS;

  // Stage z into LDS (async; 8 x b128 per thread).
  for (int g = t; g < NIN / 4; g += 256)
    async_g2l_b128(z + 4 * g, zs + 4 * g);

  auto issue_tile = [&](int c) {
    float* buf = tile[c & 1];
    for (int g = t; g < ROWS * KC / 4; g += 256) {       // 2 per thread
      int rr = g >> 5;                                   // KC/4 == 32 granules/row
      int cc = g & 31;
      async_g2l_b128(W + (size_t)(row0 + rr) * NIN + c * KC + cc * 4,
                     buf + rr * LDW + cc * 4);
    }
  };

  const int nChunks = NIN / KC;  // 64
  issue_tile(0);

  v8f acc = {0.f, 0.f, 0.f, 0.f, 0.f, 0.f, 0.f, 0.f};
  v2f ones; ones.x = 1.f; ones.y = 1.f;
  const int r16 = lane & 15;
  const int half2 = (lane >> 4) << 1;  // lanes 16-31 supply K+2,K+3 of the A tile

  for (int c = 0; c < nChunks; ++c) {
    if (c + 1 < nChunks) { issue_tile(c + 1); wait_async<2>(); }
    else                 { wait_async<0>(); }
    __syncthreads();
    const float* tl = tile[c & 1] + r16 * LDW + wave * 16 + half2;
#pragma unroll
    for (int kk = 0; kk < 16; kk += 4) {
      v2f a; a.x = tl[kk]; a.y = tl[kk + 1];
      acc = __builtin_amdgcn_wmma_f32_16x16x4_f32(
          false, a, false, ones, (short)0, acc, false, false);
    }
    __syncthreads();
  }

  // Extract row sums: column 0 of D; lane0 holds M=0..7, lane16 holds M=8..15.
  if (lane == 0) {
#pragma unroll
    for (int i = 0; i < 8; ++i) redW[wave][i] = acc[i];
  }
  if (lane == 16) {
#pragma unroll
    for (int i = 0; i < 8; ++i) redW[wave][8 + i] = acc[i];
  }
  if (t == 0) sNeedyN = 0;
  __syncthreads();

  if (t < ROWS) {
    float s = 0.f;
#pragma unroll
    for (int w = 0; w < 8; ++w) s += redW[w][t];
    sWsum[t] = s;
    sTmp[t] = s * scal[0] / (s - 1.f);   // Wsum*Zsum/(Wsum-1)
    sCnt[t] = 0;
  }
  __syncthreads();

  // L = count(z <= tmp[r]) for all 16 rows, from LDS-resident z.
  int cl[ROWS];
#pragma unroll
  for (int r = 0; r < ROWS; ++r) cl[r] = 0;
  for (int j = t; j < NIN; j += 256) {
    float zj = zs[j];
#pragma unroll
    for (int r = 0; r < ROWS; ++r) cl[r] += (zj <= sTmp[r]) ? 1 : 0;
  }
#pragma unroll
  for (int r = 0; r < ROWS; ++r) if (cl[r]) atomicAdd(&sCnt[r], cl[r]);
  __syncthreads();

  if (t < ROWS) {
    const float Wsum = sWsum[t];
    const int cnt = sCnt[t];
    const bool fc = Wsum > 1.f;
    const int jmax = ((const int*)scal)[3];
    const bool fullcase = fc ? (cnt > 0) : (cnt == 0);      // k = NIN-1
    const bool infcase  = fc ? (cnt == 0) : (cnt == NIN || cnt == 1);
    if (fullcase) {
      float Wc = Wsum - W[(size_t)(row0 + t) * NIN + jmax];
      float Zc = scal[1];                                   // sum excl. max
      out[row0 + t] = Wc * Zc / (Wc - 1.f);
    } else if (infcase) {
      out[row0 + t] = __builtin_inff();
    } else {                                                // rare: k = L-1
      int idx = atomicAdd(&sNeedyN, 1);
      sNeedy[idx] = t;
    }
  }
  __syncthreads();

  // Rare masked path: whole block per needy row, W row re-read (L2-warm).
  for (int q = 0; q < sNeedyN; ++q) {
    const int r = sNeedy[q];
    const float thr = sTmp[r];
    if (t == 0) { sMW = 0.f; sMZ = 0.f; sMM = 0ull; }
    __syncthreads();
    const float* Wr = W + (size_t)(row0 + r) * NIN;
    float pw = 0.f, pz = 0.f;
    unsigned long long pm = 0ull;
    for (int j = t; j < NIN; j += 256) {
      float zj = zs[j];
      if (zj <= thr) {
        pw += Wr[j]; pz += zj;
        unsigned b = __float_as_uint(zj);
        unsigned key = (b & 0x80000000u) ? ~b : (b | 0x80000000u);
        unsigned long long pk = ((unsigned long long)key << 32) | (unsigned)j;
        if (pk > pm) pm = pk;
      }
    }
    atomicAdd(&sMW, pw);
    atomicAdd(&sMZ, pz);
    atomicMax(&sMM, pm);
    __syncthreads();
    if (t == 0) {
      int js = (int)(sMM & 0xffffffffu);
      float Wc = sMW - Wr[js];
      float Zc = sMZ - zs[js];
      out[row0 + r] = Wc * Zc / (Wc - 1.f);
    }
    __syncthreads();
  }
}

// ============================================================================
// Host side
// ============================================================================
extern "C" void kernel_launch(void* const* d_in, const int* in_sizes, int n_in_args,
                              void* d_out, int out_size, void* d_ws, size_t ws_size,
                              hipStream_t stream) {
  (void)in_sizes; (void)n_in_args; (void)out_size; (void)ws_size;
  const float* x  = (const float*)d_in[0];
  const float* W1 = (const float*)d_in[1];   // [HID_DIM, IN_DIM]
  const float* W2 = (const float*)d_in[2];   // [OUT_DIM, HID_DIM]
  float* out = (float*)d_out;

  float* ws    = (float*)d_ws;
  float* z0    = ws;               // IN_DIM floats
  float* z1    = ws + IN_DIM;      // HID_DIM floats
  float* scal0 = ws + IN_DIM + HID_DIM;      // 4 slots
  float* scal1 = scal0 + 4;                  // 4 slots

  prep_kernel<<<1, 256, 0, stream>>>(x, z0, scal0, IN_DIM, 1);
  spiking_layer_kernel<<<HID_DIM / ROWS, 256, 0, stream>>>(W1, z0, scal0, z1);
  prep_kernel<<<1, 256, 0, stream>>>(z1, nullptr, scal1, HID_DIM, 0);
  spiking_layer_kernel<<<OUT_DIM / ROWS, 256, 0, stream>>>(W2, z1, scal1, out);
}